// PointToPixel_16999480558180
// MI455X (gfx1250) — compile-verified
//
#include <hip/hip_runtime.h>
#include <hip/hip_bf16.h>

// Point-cloud rasterization for MI455X (gfx1250).
// Memory/atomic-bound scatter: WMMA not applicable (projection "matmul" has
// K-dim 3 -> 4 scalar FMAs per point; no 16x16xK tiling can win).
// CDNA5-aware choices:
//  - TH=NT (nontemporal) b128 loads for the 96MB one-pass point/color stream,
//    keeping the 32MB img/acc accumulation surfaces resident in the 192MB L2.
//  - no-return global_atomic_add_f32 scatter (STOREcnt path, no return wait).
//  - b128 (float4) zero-init and normalize to minimize VMEM issue count.
//  - wave32-friendly 256-thread blocks.

typedef float v4f __attribute__((ext_vector_type(4)));

__global__ void ptp_zero_vec_kernel(float* __restrict__ p, long n) {
    long i = ((long)blockIdx.x * blockDim.x + threadIdx.x) * 4;
    const long stride = (long)gridDim.x * blockDim.x * 4;
    for (; i < n; i += stride) {
        if (i + 3 < n) {
            v4f z = {0.0f, 0.0f, 0.0f, 0.0f};
            *(v4f*)(p + i) = z;                 // global_store_b128
        } else {
            for (long j = i; j < n; ++j) p[j] = 0.0f;
        }
    }
}

// Per-point projection + scatter (shared by vector and scalar paths).
__device__ __forceinline__ void ptp_point(
    float px, float py, float pz, float c0, float c1, float c2,
    long b, int H, int W, long HW,
    float k00, float k01, float k02, float k10, float k11, float k12,
    float* __restrict__ img, float* __restrict__ acc) {
    const float cx = px / pz;
    const float cy = py / pz;
    const float uf = fmaf(k01, cy, fmaf(k00, cx, k02));
    const float vf = fmaf(k11, cy, fmaf(k10, cx, k12));
    const int u = (int)rintf(uf);   // round-half-to-even, matches jnp.round
    const int v = (int)rintf(vf);
    const bool cond = (u > 0) & (u < W) & (v > 0) & (v < H) & (pz > 0.0f);
    if (!cond) return;  // reference adds weight-0 contribution at ind=0 (no-op)
    // Faithful to source: ind = u + v * H (H == W in this setup)
    const long ind = (long)u + (long)v * (long)H;
    float* imgb = img + b * 3 * HW + ind;
    atomicAdd(imgb,           c0);   // global_atomic_add_f32, no return
    atomicAdd(imgb +     HW,  c1);
    atomicAdd(imgb + 2 * HW,  c2);
    atomicAdd(acc + b * HW + ind, 1.0f);
}

// ch == 3 fast path: 4 points per thread, 3x b128 NT loads per stream.
__global__ void ptp_scatter3_kernel(const float* __restrict__ x,
                                    const float* __restrict__ c,
                                    const float* __restrict__ K,
                                    const int* __restrict__ Hp,
                                    const int* __restrict__ Wp,
                                    float* __restrict__ img,   // (B, 3, H, W)
                                    float* __restrict__ acc,   // (B, H, W)
                                    long P, long out_elems) {
    const long j0 = ((long)blockIdx.x * blockDim.x + threadIdx.x) * 4;
    if (j0 >= P) return;

    const int H = *Hp;
    const int W = *Wp;
    const long HW = (long)H * (long)W;
    const long B = out_elems / (3 * HW);
    const long N = P / B;

    const float k00 = K[0], k01 = K[1], k02 = K[2];
    const float k10 = K[3], k11 = K[4], k12 = K[5];

    if (j0 + 3 < P) {
        // 4 points * 3 floats = 48B, 16B-aligned (base aligned, 48*t offset).
        const v4f xa = __builtin_nontemporal_load((const v4f*)(x + 3 * j0));
        const v4f xb = __builtin_nontemporal_load((const v4f*)(x + 3 * j0 + 4));
        const v4f xc = __builtin_nontemporal_load((const v4f*)(x + 3 * j0 + 8));
        const v4f ca = __builtin_nontemporal_load((const v4f*)(c + 3 * j0));
        const v4f cb = __builtin_nontemporal_load((const v4f*)(c + 3 * j0 + 4));
        const v4f cc = __builtin_nontemporal_load((const v4f*)(c + 3 * j0 + 8));

        const float px[4] = {xa.x, xa.w, xb.z, xc.y};
        const float py[4] = {xa.y, xb.x, xb.w, xc.z};
        const float pz[4] = {xa.z, xb.y, xc.x, xc.w};
        const float r0[4] = {ca.x, ca.w, cb.z, cc.y};
        const float r1[4] = {ca.y, cb.x, cb.w, cc.z};
        const float r2[4] = {ca.z, cb.y, cc.x, cc.w};

        // One i64 division; neighbors corrected by carry (N >= 4 always here).
        long b = j0 / N;
        long r = j0 - b * N;
#pragma unroll
        for (int k = 0; k < 4; ++k) {
            long bb = b, rr = r + k;
            if (rr >= N) { rr -= N; bb += 1; }
            ptp_point(px[k], py[k], pz[k], r0[k], r1[k], r2[k],
                      bb, H, W, HW, k00, k01, k02, k10, k11, k12, img, acc);
        }
    } else {
        for (long j = j0; j < P; ++j) {
            const long b = j / N;
            const float px = __builtin_nontemporal_load(x + 3 * j + 0);
            const float py = __builtin_nontemporal_load(x + 3 * j + 1);
            const float pz = __builtin_nontemporal_load(x + 3 * j + 2);
            const float c0 = __builtin_nontemporal_load(c + 3 * j + 0);
            const float c1 = __builtin_nontemporal_load(c + 3 * j + 1);
            const float c2 = __builtin_nontemporal_load(c + 3 * j + 2);
            ptp_point(px, py, pz, c0, c1, c2,
                      b, H, W, HW, k00, k01, k02, k10, k11, k12, img, acc);
        }
    }
}

// Generic scalar scatter (fallback for ch != 3).
__global__ void ptp_scatter_gen_kernel(const float* __restrict__ x,
                                       const float* __restrict__ c,
                                       const float* __restrict__ K,
                                       const int* __restrict__ Hp,
                                       const int* __restrict__ Wp,
                                       float* __restrict__ img,
                                       float* __restrict__ acc,
                                       long P, long out_elems, int ch) {
    const long i = (long)blockIdx.x * blockDim.x + threadIdx.x;
    if (i >= P) return;
    const int H = *Hp;
    const int W = *Wp;
    const long HW = (long)H * (long)W;
    const long B = out_elems / ((long)ch * HW);
    const long N = P / B;
    const long b = i / N;
    const float k00 = K[0], k01 = K[1], k02 = K[2];
    const float k10 = K[3], k11 = K[4], k12 = K[5];

    const float px = __builtin_nontemporal_load(x + 3 * i + 0);
    const float py = __builtin_nontemporal_load(x + 3 * i + 1);
    const float pz = __builtin_nontemporal_load(x + 3 * i + 2);
    const float cx = px / pz;
    const float cy = py / pz;
    const int u = (int)rintf(fmaf(k01, cy, fmaf(k00, cx, k02)));
    const int v = (int)rintf(fmaf(k11, cy, fmaf(k10, cx, k12)));
    const bool cond = (u > 0) & (u < W) & (v > 0) & (v < H) & (pz > 0.0f);
    if (!cond) return;
    const long ind = (long)u + (long)v * (long)H;
    float* imgb = img + b * (long)ch * HW + ind;
    const float* cp = c + i * (long)ch;
    for (int k = 0; k < ch; ++k) {
        atomicAdd(imgb + (long)k * HW, __builtin_nontemporal_load(cp + k));
    }
    atomicAdd(acc + b * HW + ind, 1.0f);
}

// Normalize: 4 pixels per thread, b128 per channel plane when possible.
__global__ void ptp_normalize_kernel(float* __restrict__ img,       // (B, ch, H, W)
                                     const float* __restrict__ acc, // (B, H, W)
                                     const int* __restrict__ Hp,
                                     const int* __restrict__ Wp,
                                     long pix_total, int ch) {
    const long p0 = ((long)blockIdx.x * blockDim.x + threadIdx.x) * 4;
    if (p0 >= pix_total) return;
    const long HW = (long)(*Hp) * (long)(*Wp);
    const long b0 = p0 / HW;
    const long b3 = (p0 + 3) / HW;

    if (((HW & 3) == 0) && (b0 == b3) && (p0 + 3 < pix_total)) {
        const v4f a = *(const v4f*)(acc + p0);     // 16B-aligned: p0%4==0
        v4f inv;
        inv.x = 1.0f / ((a.x == 0.0f) ? 1.0f : a.x);
        inv.y = 1.0f / ((a.y == 0.0f) ? 1.0f : a.y);
        inv.z = 1.0f / ((a.z == 0.0f) ? 1.0f : a.z);
        inv.w = 1.0f / ((a.w == 0.0f) ? 1.0f : a.w);
        const long pix = p0 - b0 * HW;             // %4==0 since HW%4==0
        float* base = img + b0 * (long)ch * HW + pix;
        for (int k = 0; k < ch; ++k) {
            v4f* ptr = (v4f*)(base + (long)k * HW);
            v4f val = *ptr;
            val *= inv;
            *ptr = val;
        }
    } else {
        for (long p = p0; p < pix_total && p < p0 + 4; ++p) {
            const long b = p / HW;
            const long pix = p - b * HW;
            float a = acc[p];
            a = (a == 0.0f) ? 1.0f : a;
            const float iv = 1.0f / a;
            float* base = img + b * (long)ch * HW + pix;
            for (int k = 0; k < ch; ++k) base[(long)k * HW] *= iv;
        }
    }
}

extern "C" void kernel_launch(void* const* d_in, const int* in_sizes, int n_in,
                              void* d_out, int out_size, void* d_ws, size_t ws_size,
                              hipStream_t stream) {
    const float* x  = (const float*)d_in[0];  // (B, N, 3)
    const float* c  = (const float*)d_in[1];  // (B, N, ch)
    const float* K  = (const float*)d_in[2];  // (3, 3)
    const int*   Hp = (const int*)d_in[3];    // scalar
    const int*   Wp = (const int*)d_in[4];    // scalar

    float* img = (float*)d_out;               // (B, ch, H, W)
    float* acc = (float*)d_ws;                // (B, H, W) scratch

    const long P         = (long)in_sizes[0] / 3;          // B*N points
    const int  ch        = (int)((long)in_sizes[1] / P);   // channels (3)
    const long out_elems = (long)out_size;                 // B*ch*H*W
    const long pix_total = out_elems / (long)ch;           // B*H*W

    const int threads = 256;  // 8 wave32 waves per block

    // Zero accumulation surfaces every call (harness poisons to 0xAA).
    {
        long work = (out_elems + 3) / 4;
        int blocks = (int)((work + threads - 1) / threads);
        if (blocks > 65535) blocks = 65535;  // grid-stride covers the rest
        ptp_zero_vec_kernel<<<blocks, threads, 0, stream>>>(img, out_elems);
    }
    {
        long work = (pix_total + 3) / 4;
        int blocks = (int)((work + threads - 1) / threads);
        if (blocks > 65535) blocks = 65535;
        ptp_zero_vec_kernel<<<blocks, threads, 0, stream>>>(acc, pix_total);
    }

    // Scatter.
    if (ch == 3) {
        const long nthreads_total = (P + 3) / 4;
        const int blocks = (int)((nthreads_total + threads - 1) / threads);
        ptp_scatter3_kernel<<<blocks, threads, 0, stream>>>(
            x, c, K, Hp, Wp, img, acc, P, out_elems);
    } else {
        const int blocks = (int)((P + threads - 1) / threads);
        ptp_scatter_gen_kernel<<<blocks, threads, 0, stream>>>(
            x, c, K, Hp, Wp, img, acc, P, out_elems, ch);
    }

    // Normalize in place: 4 pixels per thread across all channels.
    {
        const long nthreads_total = (pix_total + 3) / 4;
        const int blocks = (int)((nthreads_total + threads - 1) / threads);
        ptp_normalize_kernel<<<blocks, threads, 0, stream>>>(
            img, acc, Hp, Wp, pix_total, ch);
    }
}